// SpAdjDropEdge_48541720379660
// MI455X (gfx1250) — compile-verified
//
#include <hip/hip_runtime.h>
#include <hip/hip_bf16.h>

// 16-byte vector type that __builtin_nontemporal_* accepts (ext_vector, not HIP's float4 struct)
typedef __attribute__((ext_vector_type(4))) float f32x4;

#define MASK_BLOCK 256
#define CPY_BLOCK  256

// ---------------------------------------------------------------------------
// Kernel 1: new_vals = floor(rand_u + keep) ? vals * (1/keep) : 0
// Streaming, no reuse -> non-temporal B128 loads/stores.
// ---------------------------------------------------------------------------
__global__ void __launch_bounds__(MASK_BLOCK)
drop_edge_mask_kernel(const float* __restrict__ vals,
                      const float* __restrict__ rnd,
                      const float* __restrict__ keep_ptr,
                      float* __restrict__ out,
                      int n) {
    const int t  = blockIdx.x * blockDim.x + threadIdx.x;
    const int i4 = t * 4;

    // uniform scalar: compiler emits s_load; reciprocal computed once per thread
    const float keep = *keep_ptr;
    const float inv  = 1.0f / keep;

    if (i4 + 3 < n) {
        f32x4 v = __builtin_nontemporal_load((const f32x4*)(vals + i4));
        f32x4 r = __builtin_nontemporal_load((const f32x4*)(rnd  + i4));
        f32x4 o;
        o.x = (__builtin_floorf(r.x + keep) != 0.0f) ? v.x * inv : 0.0f;
        o.y = (__builtin_floorf(r.y + keep) != 0.0f) ? v.y * inv : 0.0f;
        o.z = (__builtin_floorf(r.z + keep) != 0.0f) ? v.z * inv : 0.0f;
        o.w = (__builtin_floorf(r.w + keep) != 0.0f) ? v.w * inv : 0.0f;
        __builtin_nontemporal_store(o, (f32x4*)(out + i4));
    } else {
        // scalar tail (only taken if n % 4 != 0)
        for (int j = i4; j < n; ++j) {
            float r = rnd[j];
            out[j] = (__builtin_floorf(r + keep) != 0.0f) ? vals[j] * inv : 0.0f;
        }
    }
}

// ---------------------------------------------------------------------------
// Kernel 2: pass-through copy of idxs (128 MB) via the CDNA5 async LDS path:
//   global_load_async_to_lds_b128  ->  s_wait_asynccnt 0  ->
//   global_store_async_from_lds_b128
// Double-buffered LDS (parity p) so the next iteration's async load never
// overwrites the region a still-outstanding async store is reading; one
// s_wait_asynccnt 0 per iteration then suffices (it covers both the just-
// issued load and the previous iteration's store).
// ---------------------------------------------------------------------------
__global__ void __launch_bounds__(CPY_BLOCK)
copy_idx_async_kernel(const int* __restrict__ src,
                      int* __restrict__ dst,
                      long n) {
    __shared__ int buf[2][CPY_BLOCK * 4];   // 2 x 4KB double buffer

    const unsigned tid    = threadIdx.x;
    const long nchunks    = n >> 2;         // 16B chunks
    const long stride     = (long)gridDim.x * blockDim.x;
    long c                = (long)blockIdx.x * blockDim.x + tid;
    unsigned p            = 0;

    for (; c < nchunks; c += stride) {
        // lower 32 bits of the flat shared-space pointer == LDS byte offset
        unsigned lds = (unsigned)(unsigned long long)&buf[p][tid * 4];
        unsigned long long ga = (unsigned long long)(src + 4 * c);
        unsigned long long gd = (unsigned long long)(dst + 4 * c);

        asm volatile("global_load_async_to_lds_b128 %0, %1, off"
                     :: "v"(lds), "v"(ga) : "memory");
        // wait: just-issued load complete (and prior iteration's store drained)
        asm volatile("s_wait_asynccnt 0x0" ::: "memory");
        asm volatile("global_store_async_from_lds_b128 %0, %1, off"
                     :: "v"(gd), "v"(lds) : "memory");

        p ^= 1;
    }
    // drain outstanding async stores before wave exit
    asm volatile("s_wait_asynccnt 0x0" ::: "memory");

    // scalar tail (only if n % 4 != 0)
    const long tail = n & 3;
    if (blockIdx.x == 0 && (long)tid < tail) {
        dst[n - tail + tid] = src[n - tail + tid];
    }
}

// ---------------------------------------------------------------------------
// Launch
// ---------------------------------------------------------------------------
extern "C" void kernel_launch(void* const* d_in, const int* in_sizes, int n_in,
                              void* d_out, int out_size, void* d_ws, size_t ws_size,
                              hipStream_t stream) {
    const float* vals     = (const float*)d_in[0];
    const int*   idxs     = (const int*)  d_in[1];
    const float* rand_u   = (const float*)d_in[2];
    const float* keep_ptr = (const float*)d_in[3];

    const int E    = in_sizes[0];      // 16,000,000
    const long nIdx = (long)in_sizes[1]; // 32,000,000

    float* out_vals = (float*)d_out;
    int*   out_idx  = (int*)((float*)d_out + E);  // int32 bits stored after the f32 block

    // Kernel 1: one f32x4 per thread
    {
        int nvec   = (E + 3) / 4;
        int blocks = (nvec + MASK_BLOCK - 1) / MASK_BLOCK;
        if (blocks < 1) blocks = 1;
        drop_edge_mask_kernel<<<blocks, MASK_BLOCK, 0, stream>>>(
            vals, rand_u, keep_ptr, out_vals, E);
    }

    // Kernel 2: async-LDS streaming copy, grid-stride
    {
        long chunks = nIdx >> 2;
        long want   = (chunks + CPY_BLOCK - 1) / CPY_BLOCK;
        int blocks  = (int)((want < 32768) ? (want < 1 ? 1 : want) : 32768);
        copy_idx_async_kernel<<<blocks, CPY_BLOCK, 0, stream>>>(
            idxs, out_idx, nIdx);
    }
}